// OneHotConv_34857954574522
// MI455X (gfx1250) — compile-verified
//
#include <hip/hip_runtime.h>
#include <stdint.h>

typedef float v2f __attribute__((ext_vector_type(2)));
typedef float v8f __attribute__((ext_vector_type(8)));

#define KTAPS 9
#define CIN   4
#define FILT  16
#define KC    36   // KTAPS * CIN, contraction length

// ---------------------------------------------------------------------------
// Prep 1: -softplus(decay_rate[c]) once into ws[0..3].
// ---------------------------------------------------------------------------
__global__ void softplus_prep_kernel(const float* __restrict__ decay_rate,
                                     float* __restrict__ neg_sp) {
    const int t = threadIdx.x;
    if (t < CIN) {
        const float x = decay_rate[t];
        neg_sp[t] = -(fmaxf(x, 0.0f) + log1pf(expf(-fabsf(x))));
    }
}

// ---------------------------------------------------------------------------
// Prep 2: zip[i] = { times_in[i], -softplus(decay[seg_ids[i]]) }.
// Halves the divergent gather count in the main kernel (one 8B gather per tap
// instead of two 4B gathers) and pre-resolves the per-tap decay select.
// ---------------------------------------------------------------------------
__global__ __launch_bounds__(256) void zip_prep_kernel(
    const float* __restrict__ times_in,
    const int*   __restrict__ seg_ids,
    const float* __restrict__ neg_sp,
    float2*      __restrict__ zip,
    int n_in) {
    const int i = blockIdx.x * blockDim.x + threadIdx.x;
    if (i < n_in) {
        const int c = seg_ids[i] & (CIN - 1);
        zip[i] = make_float2(times_in[i], neg_sp[c]);
    }
}

// ---------------------------------------------------------------------------
// Main kernel: grid-stride over 16-row output tiles; one wave32 per tile.
// Contraction (o,36)x(36,16) as 9 x V_WMMA_F32_16X16X4_F32 (exact f32).
// USE_ZIP=true : one float2 gather per tap (preferred).
// USE_ZIP=false: fallback, two 4B gathers + decay select (small workspace).
// ---------------------------------------------------------------------------
template <bool USE_ZIP>
__global__ __launch_bounds__(256) void onehot_conv_wmma_kernel(
    const float2* __restrict__ zip,          // (n_in,)  {t_pred, -softplus}
    const float*  __restrict__ times_in,     // (n_in,)        [fallback]
    const int*    __restrict__ seg_ids,      // (n_in,)        [fallback]
    const float*  __restrict__ times_out,    // (n_out,)
    const int*    __restrict__ pred_ids,     // (n_out, 9, 4) row-major
    const float*  __restrict__ neg_sp,       // (4,)           [fallback]
    const float*  __restrict__ kern,         // (36, 16)
    const float*  __restrict__ bias,         // (16,)
    float*        __restrict__ out,          // (n_out, 16)
    int n_in, int n_out)
{
    const int lane   = threadIdx.x & 31;
    const int wave   = threadIdx.x >> 5;
    const int nwaves = gridDim.x * (blockDim.x >> 5);
    const int wid    = blockIdx.x * (blockDim.x >> 5) + wave;

    const int half = lane >> 4;   // lane half: 0 or 1
    const int lm   = lane & 15;   // A-row within tile / B,D column (filter)

    float d0 = 0.f, d1 = 0.f, d2 = 0.f, d3 = 0.f;
    if (!USE_ZIP) { d0 = neg_sp[0]; d1 = neg_sp[1]; d2 = neg_sp[2]; d3 = neg_sp[3]; }

    const float bv = bias[lm];

    // Hoist all 9 tile-invariant B-fragments: VGPR i, lane-half h -> K = 2h+i
    v2f bfrag[KTAPS];
    #pragma unroll
    for (int j = 0; j < KTAPS; ++j) {
        #pragma unroll
        for (int i = 0; i < 2; ++i) {
            const int kc = 4 * j + 2 * half + i;
            bfrag[j][i] = kern[kc * FILT + lm];
        }
    }

    const int ntiles = (n_out + 15) >> 4;

    for (int tile = wid; tile < ntiles; tile += nwaves) {   // wave-uniform loop
        const int  obase  = tile * 16;
        const int  row    = obase + lm;
        const bool row_ok = row < n_out;
        const int  rowc   = row_ok ? row : (n_out - 1);
        const float t_out = times_out[rowc];
        const int* prow   = pred_ids + rowc * KC;

        v8f acc = {};

        #pragma unroll
        for (int j = 0; j < KTAPS; ++j) {
            // this lane's two K-slots are adjacent ints, 8B-aligned
            // (index = rowc*36 + 4j + 2*half is even) -> one NT b64 load
            const uint64_t pp = __builtin_nontemporal_load(
                (const uint64_t*)(prow + 4 * j + 2 * half));

            v2f a;
            #pragma unroll
            for (int i = 0; i < 2; ++i) {
                const int  p     = (int)(pp >> (32 * i));
                const bool valid = row_ok && (p < n_in);
                const int  pc    = (p < n_in) ? p : 0;      // safe gather index
                float w;
                if (USE_ZIP) {
                    const float2 z = zip[pc];               // ONE 8B L2 gather
                    w = valid ? __expf(z.y * (t_out - z.x)) : 0.0f;
                } else {
                    const float tp = times_in[pc];
                    const int   c  = seg_ids[pc];
                    const float nd = (c == 0) ? d0 : (c == 1) ? d1
                                   : (c == 2) ? d2 : d3;
                    w = valid ? __expf(nd * (t_out - tp)) : 0.0f;
                }
                a[i] = w;
            }
            // D = A(16x4) * B(4x16) + C, full fp32
            acc = __builtin_amdgcn_wmma_f32_16x16x4_f32(
                /*neg_a=*/false, a, /*neg_b=*/false, bfrag[j],
                /*c_mod=*/(short)0, acc, /*reuse_a=*/false, /*reuse_b=*/false);
        }

        // D layout: VGPR v, lane L -> M = v + 8*(L/16), N = L%16
        #pragma unroll
        for (int v = 0; v < 8; ++v) {
            const int m = obase + v + 8 * half;
            if (m < n_out)
                __builtin_nontemporal_store(acc[v] + bv, out + m * FILT + lm);
        }
    }
}

extern "C" void kernel_launch(void* const* d_in, const int* in_sizes, int n_in_args,
                              void* d_out, int out_size, void* d_ws, size_t ws_size,
                              hipStream_t stream) {
    const float* times_in   = (const float*)d_in[0];
    const float* times_out  = (const float*)d_in[1];
    const int*   seg_ids    = (const int*)d_in[2];
    const int*   pred_ids   = (const int*)d_in[3];
    const float* decay_rate = (const float*)d_in[4];
    const float* kern       = (const float*)d_in[5];
    const float* bias       = (const float*)d_in[6];
    float*       out        = (float*)d_out;

    const int n_in  = in_sizes[0];   // 2,000,000
    const int n_out = in_sizes[1];   // 1,000,000

    // workspace layout: [0,16) neg_softplus[4]; [256, 256+8*n_in) zip
    float*  neg_sp   = (float*)d_ws;
    float2* zip      = (float2*)((char*)d_ws + 256);
    const size_t need = 256 + (size_t)n_in * sizeof(float2);
    const bool use_zip = (ws_size >= need);

    softplus_prep_kernel<<<1, 32, 0, stream>>>(decay_rate, neg_sp);

    const int threads = 256;                         // 8 wave32 per block
    const int ntiles  = (n_out + 15) / 16;           // 62,500 16-row tiles
    const int wavesPB = threads / 32;
    int blocks = 1536;                               // persistent-ish: ~5 tiles/wave
    const int maxBlocks = (ntiles + wavesPB - 1) / wavesPB;
    if (blocks > maxBlocks) blocks = maxBlocks;

    if (use_zip) {
        zip_prep_kernel<<<(n_in + threads - 1) / threads, threads, 0, stream>>>(
            times_in, seg_ids, neg_sp, zip, n_in);
        onehot_conv_wmma_kernel<true><<<blocks, threads, 0, stream>>>(
            zip, times_in, seg_ids, times_out, pred_ids, neg_sp, kern, bias,
            out, n_in, n_out);
    } else {
        onehot_conv_wmma_kernel<false><<<blocks, threads, 0, stream>>>(
            zip, times_in, seg_ids, times_out, pred_ids, neg_sp, kern, bias,
            out, n_in, n_out);
    }
}